// SGDFilterDesign_73315091742933
// MI455X (gfx1250) — compile-verified
//
#include <hip/hip_runtime.h>

// ---------------------------------------------------------------------------
// SGD IIR filter design (16 SOS x 6 coeffs, 512 freqs, 1000 sequential steps)
// Single workgroup of 512 threads (16 wave32) resident on one WGP.
// Gradient reduction over frequencies done with V_WMMA_F32_16X16X4_F32 using
// an all-ones A matrix (layout-invariant column-sum), c(f) folded into B.
// ---------------------------------------------------------------------------

typedef __attribute__((ext_vector_type(2))) float v2f;
typedef __attribute__((ext_vector_type(8))) float v8f;

#define NF      512      // frequencies
#define NS      16       // SOS sections
#define NC      96       // parameters / gradient components (16*6)
#define XS      98       // padded row stride (floats): 8B-aligned, spreads banks
#define N_ITERS 1000
#define LR      0.1f
#define EPS     1e-8f

__global__ __launch_bounds__(512, 1)
void SGDFilterDesign_kernel(const float* __restrict__ sos_init,
                            const float* __restrict__ target_dB,
                            float* __restrict__ out) {
  extern __shared__ float sm[];
  float* sos_s = sm;                 // [96]      current SOS coefficients
  float* cbuf  = sm + 96;            // [512]     per-frequency loss weight c(f)
  float* part  = sm + 96 + 512;      // [16][96]  per-wave gradient partials
  float* X     = sm + 96 + 512 + NS * NC; // [512][98] raw grad factors per (f, comp)

  const int tid  = threadIdx.x;      // 0..511, one thread per frequency
  const int wave = tid >> 5;
  const int lane = tid & 31;
  const int f    = tid;

  // Per-frequency constants: w in [0, pi], endpoint included (linspace(0,pi,512))
  const float w  = 3.14159265358979f * (float)f / 511.0f;
  const float c1 = __cosf(w), s1 = __sinf(w);
  const float c2 = c1 * c1 - s1 * s1;   // cos(2w)
  const float s2 = 2.0f * s1 * c1;      // sin(2w)
  const float tgt = target_dB[f];

  if (tid < NC) sos_s[tid] = sos_init[tid];
  __syncthreads();

  const v2f onesA = {1.0f, 1.0f};     // all-ones 16x4 A matrix -> column sums
  float* Xrow = X + f * XS;
  const int h   = lane >> 4;          // lane-half selects K pair
  const int col = lane & 15;          // output column within a 16-wide tile

  const v8f vzero = {0.f, 0.f, 0.f, 0.f, 0.f, 0.f, 0.f, 0.f};

#pragma unroll 1
  for (int it = 0; it < N_ITERS; ++it) {
    // ------- pass 1: forward eval + raw per-(f, param) gradient factors -----
    float mag2 = 1.0f;
#pragma unroll
    for (int k = 0; k < NS; ++k) {
      const float* s = sos_s + k * 6;   // broadcast LDS read (same addr all lanes)
      float b0 = s[0], b1 = s[1], b2 = s[2];
      float a0 = s[3], a1 = s[4], a2 = s[5];
      // B(w) = b0 + b1 e^{-jw} + b2 e^{-2jw}  (and same for A)
      float Br = fmaf(b2, c2, fmaf(b1, c1, b0));
      float Bi = -fmaf(b2, s2, b1 * s1);
      float Ar = fmaf(a2, c2, fmaf(a1, c1, a0));
      float Ai = -fmaf(a2, s2, a1 * s1);
      float bb = fmaf(Br, Br, Bi * Bi);
      float aa = fmaf(Ar, Ar, Ai * Ai);
      float rB = __builtin_amdgcn_rcpf(bb);
      float rA = __builtin_amdgcn_rcpf(aa);
      mag2 *= bb * rA;                  // |H|^2 = prod |B|^2 / |A|^2
      // raw factors Re(e^{-ijw} conj(B))/|B|^2 etc. (c(f) applied later)
      float* xr = Xrow + k * 6;
      xr[0] = Br * rB;
      xr[1] = fmaf(c1, Br, -s1 * Bi) * rB;
      xr[2] = fmaf(c2, Br, -s2 * Bi) * rB;
      xr[3] = -(Ar * rA);
      xr[4] = -fmaf(c1, Ar, -s1 * Ai) * rA;
      xr[5] = -fmaf(c2, Ar, -s2 * Ai) * rA;
    }
    float mag = __builtin_amdgcn_sqrtf(mag2);
    float mp  = mag + EPS;
    float dB  = 6.02059991328f * __log2f(mp);        // 20*log10(mag+eps)
    // c(f) = (dB - tgt) * (40 / (512 ln10)) * mag / (mag + eps)
    cbuf[f] = (dB - tgt) * 0.033929344f * mag * __builtin_amdgcn_rcpf(mp);
    // X/cbuf rows are wave-private (wave reads only its own 32 frequencies);
    // same-wave LDS ordering makes them visible without a barrier.

    // ------- stage 1: WMMA reduction over this wave's 32 frequencies --------
    v8f acc[6];
#pragma unroll
    for (int t = 0; t < 6; ++t) acc[t] = vzero;

#pragma unroll
    for (int qi = 0; qi < 8; ++qi) {
      int fbase = (wave << 5) + (qi << 2);   // 4 frequencies per WMMA (K=4)
      int fx = fbase + (h ? 2 : 0);          // K slots for this lane-half
      int fy = fx + 1;
      float cx = cbuf[fx];
      float cy = cbuf[fy];
      const float* rx = X + fx * XS;
      const float* ry = X + fy * XS;
#pragma unroll
      for (int t = 0; t < 6; ++t) {
        v2f bv;
        bv.x = rx[t * 16 + col] * cx;        // weighted B operand element (K slot 0)
        bv.y = ry[t * 16 + col] * cy;        // weighted B operand element (K slot 1)
        // D = ones(16x4) * B(4x16) + C  -> every row of D = column sums
        acc[t] = __builtin_amdgcn_wmma_f32_16x16x4_f32(
            false, onesA, false, bv, (short)0, acc[t], false, false);
      }
    }

    // every D row is identical; lanes 0..15 of element 0 hold cols 0..15
    if (lane < 16) {
#pragma unroll
      for (int t = 0; t < 6; ++t)
        part[wave * NC + t * 16 + lane] = acc[t][0];
    }
    __syncthreads();

    // ------- stage 2: sum 16 wave partials + SGD update ---------------------
    if (tid < NC) {
      float g = 0.0f;
#pragma unroll
      for (int wv = 0; wv < 16; ++wv) g += part[wv * NC + tid];
      sos_s[tid] -= LR * g;
    }
    __syncthreads();
  }

  if (tid < NC) out[tid] = sos_s[tid];
}

extern "C" void kernel_launch(void* const* d_in, const int* in_sizes, int n_in,
                              void* d_out, int out_size, void* d_ws, size_t ws_size,
                              hipStream_t stream) {
  (void)in_sizes; (void)n_in; (void)out_size; (void)d_ws; (void)ws_size;
  const float* sos_init  = (const float*)d_in[0];   // [16,6] float32
  const float* target_dB = (const float*)d_in[1];   // [512]  float32
  float* out = (float*)d_out;                       // [16,6] float32

  const size_t smem_bytes =
      (size_t)(96 + 512 + NS * NC + NF * XS) * sizeof(float);  // ~205 KB < 320 KB WGP LDS

  // Opt into large dynamic LDS (CDNA5 WGP has 320 KB). Host-side, deterministic,
  // graph-capture safe (no alloc/sync).
  hipFuncSetAttribute((const void*)SGDFilterDesign_kernel,
                      hipFuncAttributeMaxDynamicSharedMemorySize,
                      (int)smem_bytes);

  SGDFilterDesign_kernel<<<dim3(1), dim3(512), smem_bytes, stream>>>(
      sos_init, target_dB, out);
}